// TripletMiner_14851996910157
// MI455X (gfx1250) — compile-verified
//
#include <hip/hip_runtime.h>
#include <hip/hip_bf16.h>
#include <math.h>

typedef __attribute__((ext_vector_type(2))) float v2f;
typedef __attribute__((ext_vector_type(8))) float v8f;

#define N_PTS 8192
#define DIM   512
#define ROWT  128          // rows per block
#define CHUNK 1024         // columns per block (grid.y)
#define NCH   (N_PTS / CHUNK)   // 8 column chunks
#define CT    128          // column subtile
#define KB    32           // K slab staged in LDS
#define LDA   36           // padded LDS stride (floats); gcd(36,64)=4 -> conflict-free 16-row reads
#define FLT_BIG 3.402823466e38f

// ---------------- row squared norms ----------------
__global__ void norms_kernel(const float* __restrict__ f, float* __restrict__ norms) {
    int row = blockIdx.x * blockDim.x + threadIdx.x;
    if (row >= N_PTS) return;
    const float4* fr = (const float4*)(f + (size_t)row * DIM);
    float s = 0.f;
#pragma unroll 4
    for (int i = 0; i < DIM / 4; ++i) {
        float4 v = fr[i];
        s += v.x * v.x + v.y * v.y + v.z * v.z + v.w * v.w;
    }
    norms[row] = s;
}

// ---------------- gram tiles via fp32 WMMA + masked row reductions ----------------
__launch_bounds__(256)
__global__ void tile_kernel(const float* __restrict__ feat,
                            const int*   __restrict__ labels,
                            const float* __restrict__ norms,
                            float* __restrict__ pMaxAll,
                            float* __restrict__ pMaxPos,
                            float* __restrict__ pMinNeg) {
    __shared__ float Alds[ROWT * LDA];
    __shared__ float Blds[CT * LDA];
    __shared__ float rowNorm[ROWT];
    __shared__ int   rowLab[ROWT];
    __shared__ float colNorm[CHUNK];
    __shared__ int   colLab[CHUNK];

    const int tid  = threadIdx.x;
    const int wave = tid >> 5;
    const int lane = tid & 31;
    const int lo   = lane & 15;   // column within 16-wide group (C/D layout)
    const int hi   = lane >> 4;   // row-half select (C/D layout)
    const int r0   = blockIdx.x * ROWT;
    const int cc0  = blockIdx.y * CHUNK;

    for (int i = tid; i < ROWT; i += 256) {
        rowNorm[i] = norms[r0 + i];
        rowLab[i]  = labels[r0 + i];
    }
    for (int i = tid; i < CHUNK; i += 256) {
        colNorm[i] = norms[cc0 + i];
        colLab[i]  = labels[cc0 + i];
    }
    __syncthreads();

    // per-lane partials: 8 rows each (row = wave*16 + v + 8*hi)
    float maxAll[8], maxPos[8], minNeg[8];
#pragma unroll
    for (int v = 0; v < 8; ++v) { maxAll[v] = 0.f; maxPos[v] = -1.f; minNeg[v] = FLT_BIG; }

    const v8f vzero = {0.f, 0.f, 0.f, 0.f, 0.f, 0.f, 0.f, 0.f};

    for (int ct = 0; ct < CHUNK / CT; ++ct) {
        const int c0 = cc0 + ct * CT;
        v8f acc[8];
#pragma unroll
        for (int g = 0; g < 8; ++g) acc[g] = vzero;

        for (int kb = 0; kb < DIM / KB; ++kb) {
            const int k0 = kb * KB;
            // stage A (rows r0..r0+127) and B (rows c0..c0+127), 128x32 floats each
            // 1024 float4 per tile, 256 threads -> 4 each
#pragma unroll
            for (int i = 0; i < 4; ++i) {
                int idx = tid + i * 256;          // 0..1023
                int row = idx >> 3;               // /8 (8 float4 per row)
                int c4  = (idx & 7) << 2;         // *4
                float4 av = *(const float4*)(feat + (size_t)(r0 + row) * DIM + k0 + c4);
                *(float4*)(&Alds[row * LDA + c4]) = av;
                float4 bv = *(const float4*)(feat + (size_t)(c0 + row) * DIM + k0 + c4);
                *(float4*)(&Blds[row * LDA + c4]) = bv;
            }
            __syncthreads();

#pragma unroll
            for (int k4 = 0; k4 < KB; k4 += 4) {
                // A fragment 16x4: lanes 0-15 -> M=lane,K=0/1 ; lanes 16-31 -> M=lane-16,K=2/3
                v2f a = *(const v2f*)(&Alds[(wave * 16 + lo) * LDA + k4 + hi * 2]);
#pragma unroll
                for (int g = 0; g < 8; ++g) {
                    v2f b = *(const v2f*)(&Blds[(g * 16 + lo) * LDA + k4 + hi * 2]);
                    acc[g] = __builtin_amdgcn_wmma_f32_16x16x4_f32(
                        false, a, false, b, (short)0, acc[g], false, false);
                }
            }
            __syncthreads();
        }

        // gram -> distance -> masked reductions for this 128-col tile
#pragma unroll
        for (int g = 0; g < 8; ++g) {
            const int ci = ct * CT + g * 16 + lo;   // column index within chunk
            const int gc = cc0 + ci;                // global column
            const float cn = colNorm[ci];
            const int   cl = colLab[ci];
#pragma unroll
            for (int v = 0; v < 8; ++v) {
                const int rl = wave * 16 + v + hi * 8;  // row within block tile
                const int gr = r0 + rl;                  // global row
                float sq = rowNorm[rl] + cn - 2.0f * acc[g][v];
                float dist = (sq > 0.f) ? sqrtf(sq) : 0.f;
                if (gr == gc) dist = 0.f;                // zeroed diagonal
                const bool same = (rowLab[rl] == cl);
                maxAll[v] = fmaxf(maxAll[v], dist);
                if (same) {
                    if (gr != gc) maxPos[v] = fmaxf(maxPos[v], dist);
                } else {
                    minNeg[v] = fminf(minNeg[v], dist);
                }
            }
        }
    }

    // cross-lane reduce within each 16-lane half (masks 1..8 stay inside the half)
#pragma unroll
    for (int v = 0; v < 8; ++v) {
        float ma = maxAll[v], mp = maxPos[v], mn = minNeg[v];
#pragma unroll
        for (int m = 1; m <= 8; m <<= 1) {
            ma = fmaxf(ma, __shfl_xor(ma, m, 32));
            mp = fmaxf(mp, __shfl_xor(mp, m, 32));
            mn = fminf(mn, __shfl_xor(mn, m, 32));
        }
        if (lo == 0) {
            const int grow = r0 + wave * 16 + v + hi * 8;
            const size_t o = (size_t)grow * NCH + blockIdx.y;
            pMaxAll[o] = ma;
            pMaxPos[o] = mp;
            pMinNeg[o] = mn;
        }
    }
}

// ---------------- combine chunk partials ----------------
__global__ void final_kernel(const float* __restrict__ pMaxAll,
                             const float* __restrict__ pMaxPos,
                             const float* __restrict__ pMinNeg,
                             float* __restrict__ out) {
    int row = blockIdx.x * blockDim.x + threadIdx.x;
    if (row >= N_PTS) return;
    float ma = 0.f, mp = -1.f, mn = FLT_BIG;
#pragma unroll
    for (int c = 0; c < NCH; ++c) {
        ma = fmaxf(ma, pMaxAll[row * NCH + c]);
        mp = fmaxf(mp, pMaxPos[row * NCH + c]);
        mn = fminf(mn, pMinNeg[row * NCH + c]);
    }
    // no positive -> axis_min = 0 (diag zero, dists >= 0); no negative -> axis_max
    out[row * 2 + 0] = (mp >= 0.f) ? mp : 0.f;
    out[row * 2 + 1] = (mn < 3.0e38f) ? mn : ma;
}

extern "C" void kernel_launch(void* const* d_in, const int* in_sizes, int n_in,
                              void* d_out, int out_size, void* d_ws, size_t ws_size,
                              hipStream_t stream) {
    (void)in_sizes; (void)n_in; (void)out_size; (void)ws_size;
    const float* feat  = (const float*)d_in[0];
    const int* labels  = (const int*)d_in[1];
    float* ws = (float*)d_ws;

    float* norms   = ws;                                 // N
    float* pMaxAll = norms + N_PTS;                      // N*NCH
    float* pMaxPos = pMaxAll + (size_t)N_PTS * NCH;      // N*NCH
    float* pMinNeg = pMaxPos + (size_t)N_PTS * NCH;      // N*NCH
    float* out = (float*)d_out;

    norms_kernel<<<N_PTS / 256, 256, 0, stream>>>(feat, norms);

    dim3 grid(N_PTS / ROWT, NCH);
    tile_kernel<<<grid, 256, 0, stream>>>(feat, labels, norms, pMaxAll, pMaxPos, pMinNeg);

    final_kernel<<<N_PTS / 256, 256, 0, stream>>>(pMaxAll, pMaxPos, pMinNeg, out);
}